// ChGKModel_85718957294304
// MI455X (gfx1250) — compile-verified
//
#include <hip/hip_runtime.h>
#include <math.h>

#define TPB 256
#define NWAVES (TPB / 32)     // wave32: 8 waves per block
#define CAP (TPB * 10)        // MAX_TEAM_SIZE = 10
#define EPSF 1e-7f

// ---------------------------------------------------------------------------
// Kernel 1: per-block sums of team_sizes (block = TPB teams), wave32 shuffles
// ---------------------------------------------------------------------------
__global__ void block_sums(const int* __restrict__ ts, int* __restrict__ partials, int B) {
    __shared__ int s[NWAVES];
    int tid = threadIdx.x;
    int b = blockIdx.x * TPB + tid;
    int v = (b < B) ? __builtin_nontemporal_load(ts + b) : 0;
    #pragma unroll
    for (int d = 16; d > 0; d >>= 1) v += __shfl_down(v, d, 32);
    if ((tid & 31) == 0) s[tid >> 5] = v;
    __syncthreads();
    if (tid == 0) {
        int t = 0;
        #pragma unroll
        for (int w = 0; w < NWAVES; ++w) t += s[w];
        partials[blockIdx.x] = t;
    }
}

// ---------------------------------------------------------------------------
// Kernel 2: single-block exclusive scan of the block partial sums (runs once,
// 15625 elements -> negligible; simple LDS Hillis-Steele per 256-chunk)
// ---------------------------------------------------------------------------
__global__ void scan_partials(int* __restrict__ partials, int n) {
    __shared__ int s[TPB];
    __shared__ int carry;
    int tid = threadIdx.x;
    if (tid == 0) carry = 0;
    __syncthreads();
    for (int base = 0; base < n; base += TPB) {
        int i = base + tid;
        int v = (i < n) ? partials[i] : 0;
        s[tid] = v;
        __syncthreads();
        for (int off = 1; off < TPB; off <<= 1) {
            int t = (tid >= off) ? s[tid - off] : 0;
            __syncthreads();
            s[tid] += t;
            __syncthreads();
        }
        int incl  = s[tid];
        int total = s[TPB - 1];
        int c     = carry;
        if (i < n) partials[i] = c + incl - v;   // exclusive prefix
        __syncthreads();
        if (tid == 0) carry = c + total;
        __syncthreads();
    }
}

// ---------------------------------------------------------------------------
// Kernel 3: fused gather -> elementwise -> segment reduce
// ---------------------------------------------------------------------------
__global__ void chgk_main(const int* __restrict__ qidx,
                          const int* __restrict__ pidx,
                          const int* __restrict__ ts,
                          const float* __restrict__ theta,
                          const float* __restrict__ bq,
                          const float* __restrict__ log_a,
                          const float* __restrict__ ts_bias,
                          const float* __restrict__ dl_scale,
                          const float* __restrict__ dl,
                          const int* __restrict__ ttype,
                          const int* __restrict__ block_base,
                          float* __restrict__ out,
                          int B) {
    __shared__ int s_wtot[NWAVES];
    __shared__ int s_idx[CAP];

    int tid  = threadIdx.x;
    int lane = tid & 31;           // gfx1250 is wave32-only
    int wave = tid >> 5;
    int b = blockIdx.x * TPB + tid;
    int sz = (b < B) ? __builtin_nontemporal_load(ts + b) : 0;

    // intra-wave inclusive scan of team sizes (wave32 shuffles, no LDS)
    int incl = sz;
    #pragma unroll
    for (int d = 1; d < 32; d <<= 1) {
        int t = __shfl_up(incl, d, 32);
        if (lane >= d) incl += t;
    }
    if (lane == 31) s_wtot[wave] = incl;
    __syncthreads();
    int waveBase = 0, count = 0;
    #pragma unroll
    for (int w = 0; w < NWAVES; ++w) {
        int wt = s_wtot[w];
        if (w < wave) waveBase += wt;
        count += wt;
    }
    int lo   = waveBase + incl - sz;     // member offset within block's run
    int base = block_base[blockIdx.x];   // global start offset of block's run

    // Stage the block's contiguous run of player indices into LDS via the
    // CDNA5 async global->LDS path (ASYNCcnt).  Low 32 bits of a generic
    // pointer to __shared__ are the LDS byte address (ISA 10.2).  Streamed
    // once -> non-temporal hint so the L2-resident gather tables survive.
    bool staged = (count <= CAP);
    if (staged) {
        for (int i = tid; i < count; i += TPB) {
            const int* g = pidx + base + i;
            unsigned lds = (unsigned)(unsigned long long)(&s_idx[i]);
            asm volatile("global_load_async_to_lds_b32 %0, %1, off th:TH_LOAD_NT"
                         :
                         : "v"(lds), "v"(g)
                         : "memory");
        }
    }

    // Overlap the async DMA with the per-question L2 gathers (LDS-independent)
    float b_i = 0.0f, a_i = 0.0f;
    if (b < B) {
        int q  = __builtin_nontemporal_load(qidx + b);
        int tt = ttype[q];
        b_i = bq[q] + dl_scale[tt] * dl[q];
        a_i = fmaxf(expf(fminf(log_a[q], 2.0f)), EPSF);
    }

    if (staged) asm volatile("s_wait_asynccnt 0" ::: "memory");
    __syncthreads();

    if (b < B) {
        float lam = 0.0f;
        for (int j = 0; j < sz; ++j) {
            int p = staged ? s_idx[lo + j] : pidx[base + lo + j];
            float th = theta[p];
            float logit = fminf(fmaxf(a_i * th - b_i, -20.0f), 20.0f);
            lam += expf(logit);
        }
        int bi = (sz < 10) ? sz : 10;
        float prob = 1.0f - expf(-lam * expf(ts_bias[bi]));
        prob = fminf(fmaxf(prob, EPSF), 1.0f - EPSF);
        __builtin_nontemporal_store(prob, out + b);
    }
}

// ---------------------------------------------------------------------------
// Launch
// ---------------------------------------------------------------------------
extern "C" void kernel_launch(void* const* d_in, const int* in_sizes, int n_in,
                              void* d_out, int out_size, void* d_ws, size_t ws_size,
                              hipStream_t stream) {
    const int*   qidx     = (const int*)  d_in[0];  // question_indices  [B]
    const int*   pidx     = (const int*)  d_in[1];  // player_indices    [T]
    const int*   ts       = (const int*)  d_in[2];  // team_sizes        [B]
    const float* theta    = (const float*)d_in[3];  // [NUM_PLAYERS]
    const float* bq       = (const float*)d_in[4];  // [NUM_QUESTIONS]
    const float* log_a    = (const float*)d_in[5];  // [NUM_QUESTIONS]
    const float* ts_bias  = (const float*)d_in[6];  // [11]
    const float* dl_scale = (const float*)d_in[7];  // [3]
    const float* dl       = (const float*)d_in[8];  // [NUM_QUESTIONS]
    const int*   ttype    = (const int*)  d_in[9];  // [NUM_QUESTIONS]
    float*       out      = (float*)d_out;

    int B = in_sizes[0];
    int nblocks = (B + TPB - 1) / TPB;
    int* partials = (int*)d_ws;   // nblocks ints of scratch

    block_sums   <<<nblocks, TPB, 0, stream>>>(ts, partials, B);
    scan_partials<<<1,       TPB, 0, stream>>>(partials, nblocks);
    chgk_main    <<<nblocks, TPB, 0, stream>>>(qidx, pidx, ts, theta, bq, log_a,
                                               ts_bias, dl_scale, dl, ttype,
                                               partials, out, B);
}